// connectedComponents_13271448945045
// MI455X (gfx1250) — compile-verified
//
#include <hip/hip_runtime.h>
#include <stdint.h>

// Problem constants from the reference: x is (16, 2, 1024, 1024) f32.
#define B_    16
#define H_    1024
#define W_    1024
#define HW_   (H_ * W_)            // 1,048,576 pixels per image (< 2^20 labels)
#define NPIX  (B_ * HW_)           // 16,777,216
#define TPB   256
#define TBL   1024                 // LDS hash table entries (block covers 1024 px)

// ---------------- union-find primitives (labels are per-image local indices) ---------

__device__ __forceinline__ int find_root(const int* __restrict__ p, int i) {
    int j = p[i];
    while (j != i) { i = j; j = p[i]; }   // parent values only ever decrease -> terminates
    return i;
}

// Komura-style atomicMin linking union. Final root of each component is its
// minimum linear index (deterministic regardless of atomic interleavings).
// AtomicMin always targets the LARGER (loser) root -> no hot-address writes.
__device__ void unite(int* p, int a, int b) {
    a = find_root(p, a);
    b = find_root(p, b);
    while (a != b) {
        if (b < a) { int t = a; a = b; b = t; }   // ensure a < b
        int old = atomicMin(&p[b], a);
        if (old == b) return;                     // successfully linked b under a
        b = find_root(p, old);                    // b had been re-linked; chase it
        a = find_root(p, a);
    }
}

// ---------------- kernel 1: fg mask + parent init + counts/best zero ------------------

__global__ void k_init(const float* __restrict__ x,
                       int* __restrict__ parent,
                       unsigned int* __restrict__ counts,
                       unsigned long long* __restrict__ best,
                       unsigned char* __restrict__ fg) {
    int t = blockIdx.x * blockDim.x + threadIdx.x;          // one thread per 4 pixels
    if (t < B_) best[t] = 0ull;                             // zero per-batch argmax keys
    long pix = (long)t * 4;
    if (pix >= NPIX) return;
    int b  = (int)(pix / HW_);
    int hw = (int)(pix - (long)b * HW_);

    const float* xb = x + (size_t)b * 2 * HW_;
    float4 x0 = *(const float4*)(xb + hw);                  // class 0, b128 load
    float4 x1 = *(const float4*)(xb + HW_ + hw);            // class 1, b128 load

    uchar4 f;
    f.x = (unsigned char)(x1.x > x0.x);                     // argmax>0  <=>  x1 > x0
    f.y = (unsigned char)(x1.y > x0.y);
    f.z = (unsigned char)(x1.z > x0.z);
    f.w = (unsigned char)(x1.w > x0.w);
    *(uchar4*)(fg + pix) = f;

    *(int4*)(parent + pix)  = make_int4(hw, hw + 1, hw + 2, hw + 3);
    *(uint4*)(counts + pix) = make_uint4(0u, 0u, 0u, 0u);
}

// ---------------- kernel 2: reduced-edge union (Playne-Stilwell, 4 px / thread) -------
// Reduced 8-connectivity edge set preserving connectivity:
//   N edge always; NW/NE only if N is bg; W only if NW is bg.

__global__ void k_merge(const unsigned char* __restrict__ fg,
                        int* __restrict__ parent) {
    int t = blockIdx.x * blockDim.x + threadIdx.x;          // quad id
    long pix = (long)t * 4;
    if (pix >= NPIX) return;
    int b  = (int)(pix / HW_);
    int hw = (int)(pix - (long)b * HW_);
    const unsigned char* f = fg + (size_t)b * HW_;

    uchar4 c4 = *(const uchar4*)(f + hw);
    unsigned char cur[4] = { c4.x, c4.y, c4.z, c4.w };
    if (!(cur[0] | cur[1] | cur[2] | cur[3])) return;

    int* p = parent + (size_t)b * HW_;
    int h  = hw >> 10;                 // / W_
    int w0 = hw & (W_ - 1);

    unsigned char lft = (w0 > 0) ? f[hw - 1] : (unsigned char)0;
    unsigned char up[6] = {0, 0, 0, 0, 0, 0};   // columns w0-1 .. w0+4 of row above
    if (h > 0) {
        int upb = hw - W_;
        uchar4 u4 = *(const uchar4*)(f + upb);
        up[1] = u4.x; up[2] = u4.y; up[3] = u4.z; up[4] = u4.w;
        if (w0 > 0)      up[0] = f[upb - 1];
        if (w0 + 4 < W_) up[5] = f[upb + 4];
    }

    #pragma unroll
    for (int i = 0; i < 4; ++i) {
        if (!cur[i]) continue;
        int q   = hw + i;
        int upq = q - W_;
        unsigned char n  = up[i + 1];
        unsigned char nw = up[i];
        unsigned char ne = up[i + 2];
        unsigned char wf = (i == 0) ? lft : cur[i - 1];

        if (n) {
            unite(p, q, upq);                         // N covers NW/NE transitively
        } else {
            if (nw) unite(p, q, upq - 1);             // NW
            if (ne) unite(p, q, upq + 1);             // NE
        }
        if (wf && !nw) unite(p, q, q - 1);            // W (skipped if NW links us)
    }
}

// ---------------- kernel 3: flatten + LDS-aggregated count + fused argmax -------------
// Per block: combine (root, cnt) pairs in an LDS hash table, flush one global
// atomicAdd per distinct root, and fold the returned running totals into a
// packed argmax key. The add completing a root's total observes the final
// count, so atomicMax over all keys == argmax over final counts.
// key = (count << 32) | (0xFFFFFFFF - label): ties -> smallest label (jnp.argmax).

__global__ void k_flatten_count_max(const unsigned char* __restrict__ fg,
                                    int* __restrict__ parent,
                                    unsigned int* __restrict__ counts,
                                    unsigned long long* __restrict__ best) {
    __shared__ int                s_root[TBL];
    __shared__ unsigned int       s_cnt[TBL];
    __shared__ unsigned long long s_wmax[TPB / 32];

    int tid = threadIdx.x;
    for (int i = tid; i < TBL; i += TPB) { s_root[i] = -1; s_cnt[i] = 0u; }
    __syncthreads();

    long pix = ((long)blockIdx.x * TPB + tid) * 4;          // block: 1024 contiguous px
    int b  = (int)(pix / HW_);                              // uniform within block
    int hw = (int)(pix - (long)b * HW_);
    int* p = parent + (size_t)b * HW_;
    unsigned int* cb = counts + (size_t)b * HW_;

    unsigned long long key = 0ull;

    uchar4 f4 = *(const uchar4*)(fg + pix);
    unsigned char f[4] = { f4.x, f4.y, f4.z, f4.w };
    if (f[0] | f[1] | f[2] | f[3]) {
        int r[4];
        #pragma unroll
        for (int i = 0; i < 4; ++i)
            r[i] = f[i] ? find_root(p, hw + i) : (hw + i);  // concurrent writes only shorten paths
        *(int4*)(p + hw) = make_int4(r[0], r[1], r[2], r[3]);

        #pragma unroll
        for (int i = 0; i < 4; ++i) {
            if (!f[i]) continue;
            bool dup = false;
            #pragma unroll
            for (int j = 0; j < i; ++j) if (f[j] && r[j] == r[i]) dup = true;
            if (dup) continue;
            unsigned int cnt = 1;
            #pragma unroll
            for (int j = i + 1; j < 4; ++j) if (f[j] && r[j] == r[i]) cnt++;

            // insert into block hash table (open addressing, LDS atomics)
            unsigned int hslot = ((unsigned int)r[i] * 0x9E3779B1u) >> 22;  // 10 bits
            bool inserted = false;
            for (int probe = 0; probe < 64; ++probe) {
                int prev = atomicCAS(&s_root[hslot], -1, r[i]);
                if (prev == -1 || prev == r[i]) {
                    atomicAdd(&s_cnt[hslot], cnt);
                    inserted = true;
                    break;
                }
                hslot = (hslot + 1) & (TBL - 1);
            }
            if (!inserted) {                                // ultra-rare fallback
                unsigned int old = atomicAdd(&cb[r[i]], cnt);
                unsigned long long k = ((unsigned long long)(old + cnt) << 32)
                                     | (unsigned long long)(0xFFFFFFFFu - (unsigned int)r[i]);
                key = (k > key) ? k : key;
            }
        }
    }
    __syncthreads();

    // flush: one global atomic per distinct root in this block
    for (int i = tid; i < TBL; i += TPB) {
        int rt = s_root[i];
        if (rt >= 0) {
            unsigned int c   = s_cnt[i];
            unsigned int old = atomicAdd(&cb[rt], c);
            unsigned long long k = ((unsigned long long)(old + c) << 32)
                                 | (unsigned long long)(0xFFFFFFFFu - (unsigned int)rt);
            key = (k > key) ? k : key;
        }
    }

    // block-wide max reduce of keys (wave32 shuffles + LDS), one atomicMax per block
    #pragma unroll
    for (int off = 16; off > 0; off >>= 1) {
        unsigned long long o = __shfl_xor(key, off, 32);
        key = (o > key) ? o : key;
    }
    if ((tid & 31) == 0) s_wmax[tid >> 5] = key;
    __syncthreads();
    if (tid == 0) {
        unsigned long long m = s_wmax[0];
        #pragma unroll
        for (int i = 1; i < TPB / 32; ++i) m = (s_wmax[i] > m) ? s_wmax[i] : m;
        if (m != 0ull) atomicMax(&best[b], m);
    }
}

// ---------------- kernel 4: apply mask, write both channels (b128 I/O) ----------------

__global__ void k_finalize(const float* __restrict__ x,
                           const int* __restrict__ parent,
                           const unsigned long long* __restrict__ best,
                           float* __restrict__ out) {
    int t = blockIdx.x * blockDim.x + threadIdx.x;          // one thread per 4 pixels
    long pix = (long)t * 4;
    if (pix >= NPIX) return;
    int b  = (int)(pix / HW_);
    int hw = (int)(pix - (long)b * HW_);

    // decode best label; if no fg at all, key==0 -> label 0xFFFFFFFF matches nothing
    unsigned int bl = 0xFFFFFFFFu - (unsigned int)(best[b] & 0xFFFFFFFFull);

    const float* xb = x + (size_t)b * 2 * HW_;
    float4 x0 = *(const float4*)(xb + hw);
    float4 x1 = *(const float4*)(xb + HW_ + hw);
    int4  pr  = *(const int4*)(parent + pix);

    bool k0 = (x1.x > x0.x) && ((unsigned int)pr.x == bl);
    bool k1 = (x1.y > x0.y) && ((unsigned int)pr.y == bl);
    bool k2 = (x1.z > x0.z) && ((unsigned int)pr.z == bl);
    bool k3 = (x1.w > x0.w) && ((unsigned int)pr.w == bl);

    float4 o0 = make_float4(k0 ? x0.x : 0.f, k1 ? x0.y : 0.f,
                            k2 ? x0.z : 0.f, k3 ? x0.w : 0.f);
    float4 o1 = make_float4(k0 ? x1.x : 0.f, k1 ? x1.y : 0.f,
                            k2 ? x1.z : 0.f, k3 ? x1.w : 0.f);

    float* ob = out + (size_t)b * 2 * HW_;
    *(float4*)(ob + hw)       = o0;
    *(float4*)(ob + HW_ + hw) = o1;
}

// ---------------- launch --------------------------------------------------------------

extern "C" void kernel_launch(void* const* d_in, const int* in_sizes, int n_in,
                              void* d_out, int out_size, void* d_ws, size_t ws_size,
                              hipStream_t stream) {
    (void)in_sizes; (void)n_in; (void)out_size; (void)ws_size;
    const float* x = (const float*)d_in[0];
    float* out     = (float*)d_out;

    // workspace layout: parent (64MB) | counts (64MB) | best (128B) | fg (16MB)
    int* parent              = (int*)d_ws;
    unsigned int* counts     = (unsigned int*)(parent + NPIX);
    unsigned long long* best = (unsigned long long*)(counts + NPIX);
    unsigned char* fg        = (unsigned char*)(best + B_);

    const int quadBlocks = (NPIX / 4) / TPB;                // 16384, exact

    k_init<<<quadBlocks, TPB, 0, stream>>>(x, parent, counts, best, fg);
    k_merge<<<quadBlocks, TPB, 0, stream>>>(fg, parent);
    k_flatten_count_max<<<quadBlocks, TPB, 0, stream>>>(fg, parent, counts, best);
    k_finalize<<<quadBlocks, TPB, 0, stream>>>(x, parent, best, out);
}